// RelationalAttentionWithPositionRelativeSymbols_55001351192726
// MI455X (gfx1250) — compile-verified
//
#include <hip/hip_runtime.h>
#include <math.h>

// Problem constants
#define B_    4
#define L_    512
#define D_    768
#define H_    12
#define HD_   64
#define R_    12
#define RP_   64
#define NP_   257     // 2*MAX_REL+1
#define NPAD_ 272     // padded to multiple of 16

typedef __attribute__((ext_vector_type(2))) float v2f;
typedef __attribute__((ext_vector_type(8))) float v8f;

__device__ __forceinline__ v2f ld2(const float* p) { return *(const v2f*)p; }

// ---------------------------------------------------------------------------
// 768x768 transpose: Wdst[n][k] = Wsrc[k][n]. LDS 32x33 tile, coalesced both
// directions. grid (24,24), 256 threads.
// ---------------------------------------------------------------------------
__global__ __launch_bounds__(256) void transpose768(
    const float* __restrict__ Wsrc, float* __restrict__ Wdst) {
  __shared__ float tile[32][33];
  const int bx = blockIdx.x * 32, by = blockIdx.y * 32;
  const int tx = threadIdx.x & 31, ty = threadIdx.x >> 5;
  #pragma unroll
  for (int r = ty; r < 32; r += 8)
    tile[r][tx] = Wsrc[(size_t)(by + r) * D_ + bx + tx];
  __syncthreads();
  #pragma unroll
  for (int r = ty; r < 32; r += 8)
    Wdst[(size_t)(bx + r) * D_ + by + tx] = tile[tx][r];
}

// ---------------------------------------------------------------------------
// GEMM with pre-transposed weights: C[M,N] = A[M,K] @ WT[N,K]ᵀ, fp32 WMMA
// 16x16x4. One wave per 16x64 tile. Software-pipelined by one K-step so the
// next step's 5 b64 fragment loads are in flight while the current 4 WMMAs
// execute. A-row loads clamped to Mclamp so a padded M grid stays in-bounds.
// ---------------------------------------------------------------------------
__global__ __launch_bounds__(32) void gemm_wmma_f32_bt(
    const float* __restrict__ A, const float* __restrict__ WT,
    float* __restrict__ C, int Mclamp, int N, int K) {
  const int lane = threadIdx.x & 31;
  const int hi = lane >> 4;          // half-wave select (K pair / M+8)
  const int nn = lane & 15;          // lane within half: N (B/C) or M (A)
  const int n0 = blockIdx.x * 64;
  const int m0 = blockIdx.y * 16;

  const int mload = min(m0 + nn, Mclamp);
  const float* arow  = A + (size_t)mload * K + 2 * hi;
  const float* btrow = WT + (size_t)(n0 + nn) * K + 2 * hi;
  const size_t cs = 16 * (size_t)K;  // 16 columns of WT

  v8f acc0 = {}, acc1 = {}, acc2 = {}, acc3 = {};
  // prologue: fragments for k0 = 0
  v2f a  = ld2(arow);
  v2f b0 = ld2(btrow);
  v2f b1 = ld2(btrow + cs);
  v2f b2 = ld2(btrow + 2 * cs);
  v2f b3 = ld2(btrow + 3 * cs);
  for (int k0 = 4; k0 < K; k0 += 4) {
    // prefetch next step while current WMMAs run
    v2f an  = ld2(arow  + k0);
    v2f b0n = ld2(btrow + k0);
    v2f b1n = ld2(btrow + cs + k0);
    v2f b2n = ld2(btrow + 2 * cs + k0);
    v2f b3n = ld2(btrow + 3 * cs + k0);
    acc0 = __builtin_amdgcn_wmma_f32_16x16x4_f32(false, a, false, b0, (short)0, acc0, false, false);
    acc1 = __builtin_amdgcn_wmma_f32_16x16x4_f32(false, a, false, b1, (short)0, acc1, false, false);
    acc2 = __builtin_amdgcn_wmma_f32_16x16x4_f32(false, a, false, b2, (short)0, acc2, false, false);
    acc3 = __builtin_amdgcn_wmma_f32_16x16x4_f32(false, a, false, b3, (short)0, acc3, false, false);
    a = an; b0 = b0n; b1 = b1n; b2 = b2n; b3 = b3n;
  }
  acc0 = __builtin_amdgcn_wmma_f32_16x16x4_f32(false, a, false, b0, (short)0, acc0, false, false);
  acc1 = __builtin_amdgcn_wmma_f32_16x16x4_f32(false, a, false, b1, (short)0, acc1, false, false);
  acc2 = __builtin_amdgcn_wmma_f32_16x16x4_f32(false, a, false, b2, (short)0, acc2, false, false);
  acc3 = __builtin_amdgcn_wmma_f32_16x16x4_f32(false, a, false, b3, (short)0, acc3, false, false);

  #pragma unroll
  for (int r = 0; r < 8; ++r) {
    const int m = m0 + r + 8 * hi;   // C layout: VGPR r -> M=r / r+8
    float* crow = C + (size_t)m * N + n0 + nn;
    crow[0]  = acc0[r];
    crow[16] = acc1[r];
    crow[32] = acc2[r];
    crow[48] = acc3[r];
  }
}

// ---------------------------------------------------------------------------
// Scores (QKᵀ * 1/sqrt(64)) + causal softmax -> attn[B,H,L,L].
// Block: (i-tile of 16 rows, h, b), 16 waves. Waves compute 16x16 score tiles
// via fp32 WMMA into LDS (skipping fully-masked causal tiles), then wave w
// softmaxes row w with wave32 shuffle reductions.
// ---------------------------------------------------------------------------
#define LDP 520
__global__ __launch_bounds__(512) void scores_softmax(
    const float* __restrict__ QA, const float* __restrict__ KA,
    float* __restrict__ ATTN) {
  __shared__ float sc[16 * LDP];
  const int i0   = blockIdx.x * 16;
  const int h    = blockIdx.y;
  const int b    = blockIdx.z;
  const int wave = threadIdx.x >> 5;
  const int lane = threadIdx.x & 31;
  const int hi = lane >> 4, nn = lane & 15;

  const float* qbase = QA + (size_t)(b * L_ + i0 + nn) * D_ + h * HD_ + 2 * hi;
  for (int jt = wave * 2; jt < wave * 2 + 2; ++jt) {
    const int j0 = jt * 16;
    if (j0 <= i0 + 15) {                                // skip fully-masked tiles
      const float* kbase = KA + (size_t)(b * L_ + j0 + nn) * D_ + h * HD_ + 2 * hi;
      v8f acc = {};
      v2f a  = ld2(qbase);
      v2f bb = ld2(kbase);                              // Bᵀ: ka row j as column
      #pragma unroll
      for (int k0 = 4; k0 < HD_; k0 += 4) {
        v2f an  = ld2(qbase + k0);
        v2f bbn = ld2(kbase + k0);
        acc = __builtin_amdgcn_wmma_f32_16x16x4_f32(false, a, false, bb, (short)0, acc, false, false);
        a = an; bb = bbn;
      }
      acc = __builtin_amdgcn_wmma_f32_16x16x4_f32(false, a, false, bb, (short)0, acc, false, false);
      #pragma unroll
      for (int r = 0; r < 8; ++r)
        sc[(r + 8 * hi) * LDP + j0 + nn] = acc[r] * 0.125f;
    }
  }
  __syncthreads();

  // softmax: wave w -> row w (16 waves, 16 rows)
  const int row = wave;
  const int ig  = i0 + row;
  float vals[16];
  float mx = -__builtin_inff();
  #pragma unroll
  for (int t = 0; t < 16; ++t) {
    const int j = lane + 32 * t;
    const float v = (j <= ig) ? sc[row * LDP + j] : -__builtin_inff();
    vals[t] = v;
    mx = fmaxf(mx, v);
  }
  for (int off = 16; off; off >>= 1) mx = fmaxf(mx, __shfl_xor(mx, off, 32));
  float sum = 0.0f;
  #pragma unroll
  for (int t = 0; t < 16; ++t) { const float e = __expf(vals[t] - mx); vals[t] = e; sum += e; }
  for (int off = 16; off; off >>= 1) sum += __shfl_xor(sum, off, 32);
  const float inv = 1.0f / sum;
  float* arow = ATTN + ((size_t)(b * H_ + h) * L_ + ig) * L_;
  #pragma unroll
  for (int t = 0; t < 16; ++t) {
    const int j = lane + 32 * t;
    arow[j] = (j <= ig) ? vals[t] * inv : 0.0f;
  }
}

// ---------------------------------------------------------------------------
// relations[b,i,j,r] = (qr[b,i,r,:] . kr[b,j,r,:]) / sqrt(64).
// One wave per (16x16 ij-tile, b, r); fp32 WMMA over K=64, software-pipelined;
// strided scatter into the [B,L,L,R] output layout.
// ---------------------------------------------------------------------------
__global__ __launch_bounds__(32) void relations_kernel(
    const float* __restrict__ QR, const float* __restrict__ KR,
    float* __restrict__ REL) {
  const int lane = threadIdx.x & 31;
  const int hi = lane >> 4, nn = lane & 15;
  const int j0 = blockIdx.x * 16;
  const int i0 = blockIdx.y * 16;
  const int b  = blockIdx.z / R_;
  const int r  = blockIdx.z % R_;

  const float* qbase = QR + (size_t)(b * L_ + i0 + nn) * D_ + r * RP_ + 2 * hi;
  const float* kbase = KR + (size_t)(b * L_ + j0 + nn) * D_ + r * RP_ + 2 * hi;
  v8f acc = {};
  v2f a  = ld2(qbase);
  v2f bb = ld2(kbase);
  #pragma unroll
  for (int k0 = 4; k0 < RP_; k0 += 4) {
    v2f an  = ld2(qbase + k0);
    v2f bbn = ld2(kbase + k0);
    acc = __builtin_amdgcn_wmma_f32_16x16x4_f32(false, a, false, bb, (short)0, acc, false, false);
    a = an; bb = bbn;
  }
  acc = __builtin_amdgcn_wmma_f32_16x16x4_f32(false, a, false, bb, (short)0, acc, false, false);
  #pragma unroll
  for (int rr = 0; rr < 8; ++rr) {
    const int i = i0 + rr + 8 * hi;
    const int j = j0 + nn;
    REL[((size_t)(b * L_ + i) * L_ + j) * R_ + r] = acc[rr] * 0.125f;
  }
}

// ---------------------------------------------------------------------------
// attended[b,i,:] = sum_j attn[b,h,i,j] * sv[clip(j-i)+128][h,d]
//                 + sum_r (sum_j attn[b,h,i,j] * rel[b,i,j,r]) * wr[h,d,r]
// Diagonal-banded gather -> VALU kernel. Per (b,i) block: stage attn rows and
// the rel slice in LDS, reduce ar[h][r], then accumulate over j<=i (causal).
// ---------------------------------------------------------------------------
__global__ __launch_bounds__(256) void attend_kernel(
    const float* __restrict__ ATTN, const float* __restrict__ REL,
    const float* __restrict__ SV, const float* __restrict__ WR,
    float* __restrict__ OUT) {
  __shared__ float attnL[H_ * L_];   // 12 x 512
  __shared__ float relL[L_ * R_];    // 512 x 12
  __shared__ float ar[H_ * R_];      // 12 x 12
  const int i = blockIdx.x;
  const int b = blockIdx.y;
  const int t = threadIdx.x;

  for (int idx = t; idx < H_ * L_; idx += 256) {
    const int h = idx / L_, j = idx - h * L_;
    attnL[idx] = ATTN[((size_t)(b * H_ + h) * L_ + i) * L_ + j];
  }
  const float* relrow = REL + (size_t)(b * L_ + i) * L_ * R_;
  for (int idx = t; idx < L_ * R_; idx += 256) relL[idx] = relrow[idx];
  __syncthreads();

  if (t < H_ * R_) {
    const int h = t / R_, r = t - h * R_;
    float s = 0.0f;
    for (int j = 0; j <= i; ++j) s += attnL[h * L_ + j] * relL[j * R_ + r];
    ar[t] = s;
  }
  __syncthreads();

  float acc[3] = {0.0f, 0.0f, 0.0f};
  for (int j = 0; j <= i; ++j) {
    int rp = j - i + 128;            // j<=i (causal) -> rp in [?,128]
    if (rp < 0) rp = 0;
    const float* svrow = SV + (size_t)rp * D_;
    #pragma unroll
    for (int q = 0; q < 3; ++q) {
      const int dd = t + 256 * q;
      acc[q] += attnL[(dd >> 6) * L_ + j] * svrow[dd];
    }
  }
  float* orow = OUT + (size_t)(b * L_ + i) * D_;
  #pragma unroll
  for (int q = 0; q < 3; ++q) {
    const int dd = t + 256 * q;
    const int h = dd >> 6, d = dd & 63;
    float s = acc[q];
    #pragma unroll
    for (int r = 0; r < R_; ++r) s += ar[h * R_ + r] * WR[(h * HD_ + d) * R_ + r];
    orow[dd] = s;
  }
}

// ---------------------------------------------------------------------------
extern "C" void kernel_launch(void* const* d_in, const int* in_sizes, int n_in,
                              void* d_out, int out_size, void* d_ws, size_t ws_size,
                              hipStream_t stream) {
  const float* x       = (const float*)d_in[0];
  const float* wq_attn = (const float*)d_in[1];
  const float* wk_attn = (const float*)d_in[2];
  const float* wq_rel  = (const float*)d_in[3];
  const float* wk_rel  = (const float*)d_in[4];
  const float* wr      = (const float*)d_in[5];
  const float* wv      = (const float*)d_in[6];
  const float* wo      = (const float*)d_in[7];
  const float* rel_emb = (const float*)d_in[8];

  float* out  = (float*)d_out;                                  // [B,L,768]
  float* attn = out + (size_t)B_ * L_ * D_;                     // [B,H,L,L]
  float* rel  = attn + (size_t)B_ * H_ * L_ * L_;               // [B,L,L,R]

  const size_t NBLD = (size_t)B_ * L_ * D_;
  float* qa  = (float*)d_ws;
  float* ka  = qa + NBLD;
  float* qr  = ka + NBLD;
  float* kr  = qr + NBLD;
  float* sv  = kr + NBLD;                                       // [272,768]
  float* att = sv + (size_t)NPAD_ * D_;                         // [B,L,768]
  float* wt  = att + NBLD;                                      // [768,768] scratch

  const dim3 b32(32);
  const dim3 tb(256);
  const dim3 gT(D_ / 32, D_ / 32);                              // (24,24)
  const dim3 gProj(D_ / 64, (B_ * L_) / 16);                    // (12,128)

  // Projections: qa/ka/qr/kr = x @ W  (M=2048, K=N=768), weights transposed
  // into wt first; stream order serializes reuse of wt.
  transpose768<<<gT, tb, 0, stream>>>(wq_attn, wt);
  gemm_wmma_f32_bt<<<gProj, b32, 0, stream>>>(x, wt, qa, B_ * L_ - 1, D_, D_);
  transpose768<<<gT, tb, 0, stream>>>(wk_attn, wt);
  gemm_wmma_f32_bt<<<gProj, b32, 0, stream>>>(x, wt, ka, B_ * L_ - 1, D_, D_);
  transpose768<<<gT, tb, 0, stream>>>(wq_rel, wt);
  gemm_wmma_f32_bt<<<gProj, b32, 0, stream>>>(x, wt, qr, B_ * L_ - 1, D_, D_);
  transpose768<<<gT, tb, 0, stream>>>(wk_rel, wt);
  gemm_wmma_f32_bt<<<gProj, b32, 0, stream>>>(x, wt, kr, B_ * L_ - 1, D_, D_);
  // sv = rel_emb @ wv (M=257 padded to 272, row-clamped loads)
  transpose768<<<gT, tb, 0, stream>>>(wv, wt);
  gemm_wmma_f32_bt<<<dim3(D_ / 64, NPAD_ / 16), b32, 0, stream>>>(rel_emb, wt, sv, NP_ - 1, D_, D_);

  // attn: causal softmax(QKᵀ/8)
  scores_softmax<<<dim3(L_ / 16, H_, B_), 512, 0, stream>>>(qa, ka, attn);

  // relations: full (non-causal) output
  relations_kernel<<<dim3(L_ / 16, L_ / 16, B_ * R_), b32, 0, stream>>>(qr, kr, rel);

  // attended symbols + relations -> att [B,L,768]
  attend_kernel<<<dim3(L_, B_), 256, 0, stream>>>(attn, rel, sv, wr, att);

  // out = att @ wo
  transpose768<<<gT, tb, 0, stream>>>(wo, wt);
  gemm_wmma_f32_bt<<<gProj, b32, 0, stream>>>(att, wt, out, B_ * L_ - 1, D_, D_);
}